// Sparsemax_80152679678110
// MI455X (gfx1250) — compile-verified
//
#include <hip/hip_runtime.h>
#include <stdint.h>

typedef __attribute__((ext_vector_type(4))) unsigned int v4u;
typedef __attribute__((ext_vector_type(8))) int v8i;
typedef __attribute__((ext_vector_type(4))) int v4i;

#define ROWLEN 2048
#define THREADS 256
#define PER_THREAD (ROWLEN / THREADS)  // 8
#define NWAVES (THREADS / 32)          // 8

__device__ __forceinline__ float waveReduceSum(float v) {
#pragma unroll
  for (int m = 16; m >= 1; m >>= 1) v += __shfl_xor(v, m, 32);
  return v;
}

__device__ __forceinline__ float waveReduceMax(float v) {
#pragma unroll
  for (int m = 16; m >= 1; m >>= 1) v = fmaxf(v, __shfl_xor(v, m, 32));
  return v;
}

__global__ __launch_bounds__(THREADS) void sparsemax_rows_kernel(
    const float* __restrict__ x, float* __restrict__ y) {
  __shared__ __align__(16) float lds_row[ROWLEN];
  __shared__ float red[NWAVES];
  __shared__ float red2[NWAVES];

  const int tid = threadIdx.x;
  const int lane = tid & 31;
  const int wave = tid >> 5;
  const size_t row = blockIdx.x;
  const float* __restrict__ xrow = x + row * (size_t)ROWLEN;
  float* __restrict__ yrow = y + row * (size_t)ROWLEN;

  // ---- Stage the row into LDS via the Tensor Data Mover (CDNA5 TDM) ----
#if __has_builtin(__builtin_amdgcn_tensor_load_to_lds) && \
    __has_builtin(__builtin_amdgcn_s_wait_tensorcnt)
  if (wave == 0) {  // one wave issues the DMA (uniform branch; EXEC ignored by TDM)
    const uint32_t lds_off = (uint32_t)(uintptr_t)(void*)&lds_row[0];
    const uint64_t ga = (uint64_t)(uintptr_t)xrow;
    // D# group 0: count=1 | lds_addr | global_addr[56:0] | type=2
    v4u g0;
    g0[0] = 1u;                                                   // count=1, user mode
    g0[1] = lds_off;                                              // lds_addr (bytes)
    g0[2] = (uint32_t)(ga & 0xffffffffull);                       // global_addr[31:0]
    g0[3] = (uint32_t)((ga >> 32) & 0x01ffffffull) | (2u << 30);  // addr[56:32] | type=2
    // D# group 1: data_size=4B, tensor 2048x1, tile 2048x1, dim0 stride 2048
    v8i g1;
    g1[0] = (int)(2u << 16);                      // data_size=2 (4 bytes); mask/flags 0
    g1[1] = (int)(((uint32_t)ROWLEN & 0xffffu) << 16);     // tensor_dim0[15:0] at bits 63:48
    g1[2] = (int)(((uint32_t)ROWLEN >> 16) | (1u << 16));  // tensor_dim0[31:16] | tensor_dim1=1
    g1[3] = (int)((uint32_t)ROWLEN << 16);        // tile_dim0 = 2048 at bits 127:112
    g1[4] = 1;                                    // tile_dim1 = 1, tile_dim2 = 0
    g1[5] = ROWLEN;                               // tensor_dim0_stride[31:0] = 2048
    g1[6] = 0;                                    // stride0[47:32]=0, stride1[15:0]=0
    g1[7] = 0;                                    // stride1[47:16]=0
    v4i gz4 = {0, 0, 0, 0};                       // groups 2/3 unused (2D tile)
    v8i gz8 = {0, 0, 0, 0, 0, 0, 0, 0};          // extra group (clang-23 6-arg form)
    __builtin_amdgcn_tensor_load_to_lds(g0, g1, gz4, gz4, gz8, 0);
    __builtin_amdgcn_s_wait_tensorcnt(0);
  }
  __syncthreads();
#else
  {
    float4* dstv = (float4*)lds_row;
    const float4* srcv = (const float4*)xrow;
#pragma unroll
    for (int i = 0; i < PER_THREAD / 4; ++i)
      dstv[tid + i * THREADS] = srcv[tid + i * THREADS];
  }
  __syncthreads();
#endif

  // ---- Row data -> registers (8 floats per lane, contiguous ownership) ----
  float z[PER_THREAD];
  {
    const float4* v = (const float4*)&lds_row[tid * PER_THREAD];
    float4 a = v[0], b = v[1];
    z[0] = a.x; z[1] = a.y; z[2] = a.z; z[3] = a.w;
    z[4] = b.x; z[5] = b.y; z[6] = b.z; z[7] = b.w;
  }

  // ---- Block max reduction ----
  float m = z[0];
#pragma unroll
  for (int i = 1; i < PER_THREAD; ++i) m = fmaxf(m, z[i]);
  m = waveReduceMax(m);
  if (lane == 0) red[wave] = m;
  __syncthreads();
  float zmax = red[0];
#pragma unroll
  for (int w = 1; w < NWAVES; ++w) zmax = fmaxf(zmax, red[w]);
  __syncthreads();

  // ---- Bisection on g(tau) = sum(max(z - tau, 0)) - 1, tau in [max-1, max] ----
  float lo = zmax - 1.0f;
  float hi = zmax;
#pragma unroll 1
  for (int it = 0; it < 24; ++it) {
    const float mid = 0.5f * (lo + hi);
    float p = 0.0f;
#pragma unroll
    for (int i = 0; i < PER_THREAD; ++i) p += fmaxf(z[i] - mid, 0.0f);
    p = waveReduceSum(p);
    if (lane == 0) red[wave] = p;
    __syncthreads();
    float g = -1.0f;
#pragma unroll
    for (int w = 0; w < NWAVES; ++w) g += red[w];
    __syncthreads();
    const bool c = (g >= 0.0f);   // uniform across block (identical arithmetic)
    lo = c ? mid : lo;
    hi = c ? hi : mid;
  }

  // ---- Exact polish: support stats at lo, tau = (sum_support - 1) / k ----
  float cs = 0.0f, cc = 0.0f;
#pragma unroll
  for (int i = 0; i < PER_THREAD; ++i) {
    const bool in = (z[i] > lo);
    cs += in ? z[i] : 0.0f;
    cc += in ? 1.0f : 0.0f;
  }
  cs = waveReduceSum(cs);
  cc = waveReduceSum(cc);
  if (lane == 0) { red[wave] = cs; red2[wave] = cc; }
  __syncthreads();
  float s = 0.0f, k = 0.0f;
#pragma unroll
  for (int w = 0; w < NWAVES; ++w) { s += red[w]; k += red2[w]; }
  const float tau = (s - 1.0f) / k;  // k >= 1 guaranteed (zmax > lo always)

  // ---- Output: max(z - tau, 0), coalesced b128 stores ----
  float4 o0, o1;
  o0.x = fmaxf(z[0] - tau, 0.0f);
  o0.y = fmaxf(z[1] - tau, 0.0f);
  o0.z = fmaxf(z[2] - tau, 0.0f);
  o0.w = fmaxf(z[3] - tau, 0.0f);
  o1.x = fmaxf(z[4] - tau, 0.0f);
  o1.y = fmaxf(z[5] - tau, 0.0f);
  o1.z = fmaxf(z[6] - tau, 0.0f);
  o1.w = fmaxf(z[7] - tau, 0.0f);
  float4* outv = (float4*)&yrow[tid * PER_THREAD];
  outv[0] = o0;
  outv[1] = o1;
}

extern "C" void kernel_launch(void* const* d_in, const int* in_sizes, int n_in,
                              void* d_out, int out_size, void* d_ws, size_t ws_size,
                              hipStream_t stream) {
  const float* x = (const float*)d_in[0];
  float* y = (float*)d_out;
  const int nrows = in_sizes[0] / ROWLEN;  // 2*16*2048 = 65536 rows of 2048
  sparsemax_rows_kernel<<<dim3(nrows), dim3(THREADS), 0, stream>>>(x, y);
}